// GAT_qsar_31885837206123
// MI455X (gfx1250) — compile-verified
//
#include <hip/hip_runtime.h>
#include <math.h>

typedef float v2f __attribute__((ext_vector_type(2)));
typedef float v8f __attribute__((ext_vector_type(8)));

#define NEG_SLOPE 0.2f
#define NUM_GRAPHS 2000
#define NEG_INF_KEY ((int)0x807FFFFF)   // order-key of -inf

// ---------------- order-preserving float<->int key for atomicMax -------------
__device__ __forceinline__ int f2key(float f) {
    int i = __float_as_int(f);
    return (i >= 0) ? i : (i ^ 0x7FFFFFFF);
}
__device__ __forceinline__ float key2f(int k) {
    return __int_as_float((k >= 0) ? k : (k ^ 0x7FFFFFFF));
}

// ---------------- fills (deterministic re-init each call) --------------------
__global__ void fill_f32(float* __restrict__ p, float v, size_t n) {
    size_t i = (size_t)blockIdx.x * blockDim.x + threadIdx.x;
    size_t s = (size_t)gridDim.x * blockDim.x;
    for (; i < n; i += s) p[i] = v;
}
__global__ void fill_i32(int* __restrict__ p, int v, size_t n) {
    size_t i = (size_t)blockIdx.x * blockDim.x + threadIdx.x;
    size_t s = (size_t)gridDim.x * blockDim.x;
    for (; i < n; i += s) p[i] = v;
}

// ---------------- WMMA linear: OUT[Npad x DPAD] = X[N x K] @ W[D x K]^T ------
// Both operands staged zero-padded in LDS: hot loop is ds_load + v_wmma only.
// Block = 4 waves = 64-row stripe; wave w owns rows 16w..16w+15.
// OUT has Npad = gridDim.x*64 rows reserved, so stores need NO bounds guard.
template <int KPAD, int DPAD>
__global__ void wmma_linear_kernel(const float* __restrict__ X, int ldx, int K,
                                   const float* __restrict__ W, int D,
                                   float* __restrict__ OUT, int N)
{
    __shared__ float sW[DPAD * KPAD];        // zero-padded weights
    __shared__ float sX[64 * KPAD];          // zero-padded 64-row A stripe
    const int tid  = threadIdx.x;
    const int row0 = blockIdx.x * 64;

    for (int i = tid; i < DPAD * KPAD; i += blockDim.x) {
        int n = i / KPAD, k = i % KPAD;
        sW[i] = (n < D && k < K) ? W[n * K + k] : 0.0f;
    }
    for (int i = tid; i < 64 * KPAD; i += blockDim.x) {
        int r = i / KPAD, k = i % KPAD;
        int gr = row0 + r;
        sX[i] = (gr < N && k < K) ? X[(size_t)gr * ldx + k] : 0.0f;
    }
    __syncthreads();

    const int wave  = tid >> 5;
    const int lane  = tid & 31;
    const int mlane = lane & 15;
    const int khalf = (lane >> 4) << 1;      // 0 or 2 (ISA A/B fragment layout)
    const float* xrow = sX + (wave * 16 + mlane) * KPAD;

    // base address for this lane's stores: row rbase, col mlane (tn adds 16 cols)
    const int rbase = row0 + wave * 16 + ((lane >> 4) << 3);
    float* outp = OUT + (size_t)rbase * DPAD + mlane;

    #pragma unroll
    for (int tn = 0; tn < (DPAD >> 4); ++tn) {
        const float* wrow = sW + (tn * 16 + mlane) * KPAD;
        v8f c = {0.f, 0.f, 0.f, 0.f, 0.f, 0.f, 0.f, 0.f};
        #pragma unroll
        for (int k0 = 0; k0 < KPAD; k0 += 4) {
            const int kk = k0 + khalf;
            v2f a, b;
            a.x = xrow[kk];
            a.y = xrow[kk + 1];
            b.x = wrow[kk];
            b.y = wrow[kk + 1];
            c = __builtin_amdgcn_wmma_f32_16x16x4_f32(
                    false, a, false, b, (short)0, c, false, false);
        }
        #pragma unroll
        for (int v = 0; v < 8; ++v)          // unguarded: OUT rows padded to 64
            outp[(size_t)v * DPAD + tn * 16] = c[v];
    }
}

// ---------------- edge pass 1: score + segment-max ---------------------------
__global__ void edge_score_kernel(const float* __restrict__ xl,
                                  const float* __restrict__ xr, int S,
                                  const float* __restrict__ att, int D,
                                  const int* __restrict__ src_a,
                                  const int* __restrict__ dst_a,
                                  int E, int ET,
                                  float* __restrict__ e_out,
                                  int* __restrict__ emax)
{
    int eid = blockIdx.x * blockDim.x + threadIdx.x;
    if (eid >= ET) return;
    int s, d;
    if (eid < E) { s = src_a[eid]; d = dst_a[eid]; }
    else         { s = d = eid - E; }                   // implicit self-loops
    const float* pl = xl + (size_t)s * S;
    const float* pr = xr + (size_t)d * S;
    float acc = 0.f;
    int f = 0;
    for (; f + 3 < D; f += 4) {                         // rows 16B-aligned -> b128
        float4 l  = *(const float4*)(pl + f);
        float4 r  = *(const float4*)(pr + f);
        float4 a4 = *(const float4*)(att + f);
        float m;
        m = l.x + r.x; acc += ((m > 0.f) ? m : NEG_SLOPE * m) * a4.x;
        m = l.y + r.y; acc += ((m > 0.f) ? m : NEG_SLOPE * m) * a4.y;
        m = l.z + r.z; acc += ((m > 0.f) ? m : NEG_SLOPE * m) * a4.z;
        m = l.w + r.w; acc += ((m > 0.f) ? m : NEG_SLOPE * m) * a4.w;
    }
    for (; f < D; ++f) {
        float m = pl[f] + pr[f];
        acc += ((m > 0.f) ? m : NEG_SLOPE * m) * att[f];
    }
    e_out[eid] = acc;
    atomicMax(&emax[d], f2key(acc));
}

// ---------------- edge pass 2: exp + denom + weighted scatter ----------------
// alpha = p/denom folded into node epilogue: hout accumulates p*xl, denom separately.
__global__ void edge_aggregate_kernel(const float* __restrict__ xl, int S,
                                      const int* __restrict__ src_a,
                                      const int* __restrict__ dst_a,
                                      int E, int ET,
                                      const float* __restrict__ e_in,
                                      const int* __restrict__ emax,
                                      float* __restrict__ denom,
                                      float* __restrict__ hout, int D)
{
    int eid = blockIdx.x * blockDim.x + threadIdx.x;
    if (eid >= ET) return;
    int s, d;
    if (eid < E) { s = src_a[eid]; d = dst_a[eid]; }
    else         { s = d = eid - E; }
    float em = key2f(emax[d]);
    if (!isfinite(em)) em = 0.f;                        // ref: where(isfinite, emax, 0)
    float p = __expf(e_in[eid] - em);
    atomicAdd(&denom[d], p);
    const float* pl = xl + (size_t)s * S;
    float* po = hout + (size_t)d * S;
    int f = 0;
    for (; f + 3 < D; f += 4) {
        float4 l = *(const float4*)(pl + f);            // one b128 gather
        atomicAdd(&po[f],     p * l.x);                 // scatter stays 32-bit
        atomicAdd(&po[f + 1], p * l.y);
        atomicAdd(&po[f + 2], p * l.z);
        atomicAdd(&po[f + 3], p * l.w);
    }
    for (; f < D; ++f) atomicAdd(&po[f], p * pl[f]);
}

// ---------------- node epilogue: /denom + bias (+relu) -----------------------
__global__ void node_finish_kernel(float* __restrict__ h, int S, int D,
                                   const float* __restrict__ b,
                                   const float* __restrict__ denom,
                                   int N, int do_relu)
{
    int idx = blockIdx.x * blockDim.x + threadIdx.x;
    if (idx >= N * D) return;
    int n = idx / D, f = idx - n * D;
    float dv = fmaxf(denom[n], 1e-16f);
    float v = h[(size_t)n * S + f] / dv + b[f];
    if (do_relu) v = fmaxf(v, 0.f);
    h[(size_t)n * S + f] = v;
}

// ---------------- global mean pool -------------------------------------------
__global__ void pool_kernel(const float* __restrict__ h, int S, int D,
                            const int* __restrict__ batch, int N,
                            float* __restrict__ pool, float* __restrict__ cnt)
{
    int n = blockIdx.x * blockDim.x + threadIdx.x;
    if (n >= N) return;
    int g = batch[n];
    atomicAdd(&cnt[g], 1.0f);
    const float* pn = h + (size_t)n * S;
    for (int f = 0; f < D; ++f) atomicAdd(&pool[g * D + f], pn[f]);
}

// ---------------- per-graph MLP head -----------------------------------------
__global__ void mlp_kernel(const float* __restrict__ pool,
                           const float* __restrict__ cnt,
                           const float* __restrict__ Wfc1,
                           const float* __restrict__ bfc1,
                           const float* __restrict__ Wfc2,
                           const float* __restrict__ bfc2,
                           float* __restrict__ out, int B)
{
    int g = blockIdx.x * blockDim.x + threadIdx.x;
    if (g >= B) return;
    float c = fmaxf(cnt[g], 1.0f);
    float feat[30];
    #pragma unroll
    for (int f = 0; f < 30; ++f) feat[f] = pool[g * 30 + f] / c;
    float r = bfc2[0];
    #pragma unroll
    for (int o = 0; o < 20; ++o) {
        float a = bfc1[o];
        #pragma unroll
        for (int f = 0; f < 30; ++f) a += Wfc1[o * 30 + f] * feat[f];
        r += Wfc2[o] * fmaxf(a, 0.f);
    }
    out[g] = r;
}

// =============================================================================
extern "C" void kernel_launch(void* const* d_in, const int* in_sizes, int n_in,
                              void* d_out, int out_size, void* d_ws, size_t ws_size,
                              hipStream_t stream)
{
    const float* x     = (const float*)d_in[0];
    const int*   eidx  = (const int*)  d_in[1];
    const int*   batch = (const int*)  d_in[2];
    const float* Wl[3]  = {(const float*)d_in[3],  (const float*)d_in[7],  (const float*)d_in[11]};
    const float* Wr[3]  = {(const float*)d_in[4],  (const float*)d_in[8],  (const float*)d_in[12]};
    const float* att[3] = {(const float*)d_in[5],  (const float*)d_in[9],  (const float*)d_in[13]};
    const float* bia[3] = {(const float*)d_in[6],  (const float*)d_in[10], (const float*)d_in[14]};
    const float* Wfc1 = (const float*)d_in[15];
    const float* bfc1 = (const float*)d_in[16];
    const float* Wfc2 = (const float*)d_in[17];
    const float* bfc2 = (const float*)d_in[18];

    const int N    = in_sizes[0] / 27;
    const int E    = in_sizes[1] / 2;
    const int ET   = E + N;                     // + self loops
    const int Npad = ((N + 63) / 64) * 64;      // GEMM store padding (64-row stripes)
    const int* srcA = eidx;
    const int* dstA = eidx + E;

    // layer configs: K(in dim), Dout, padded stride of output
    const int Kd[3] = {27, 100, 60};
    const int Dd[3] = {100, 60, 30};
    const int Sp[3] = {112, 64, 32};            // Dout padded to mult of 16

    // ---- workspace partition (f32 units); xl/xr padded to Npad rows ----
    float* ws = (float*)d_ws;
    size_t o = 0;
    float* xl   = ws + o; o += (size_t)Npad * 112;
    float* xr   = ws + o; o += (size_t)Npad * 112;
    float* h1   = ws + o; o += (size_t)N * 112;
    float* h2   = ws + o; o += (size_t)N * 64;
    float* h3   = ws + o; o += (size_t)N * 32;
    float* ebuf = ws + o; o += (size_t)ET;
    float* denom= ws + o; o += (size_t)N;
    int*   emax = (int*)(ws + o); o += (size_t)N;
    float* pool = ws + o; o += (size_t)NUM_GRAPHS * 30;
    float* cnt  = ws + o; o += (size_t)NUM_GRAPHS;

    const float* lin[3] = {x, h1, h2};
    const int    ldx[3] = {27, 112, 64};
    float*       hout[3]= {h1, h2, h3};

    const int gemmBlocks = Npad / 64;           // 64-row stripe per block
    const int eblk       = (ET + 255) / 256;

    for (int L = 0; L < 3; ++L) {
        const int K = Kd[L], D = Dd[L], S = Sp[L];

        // deterministic re-init of accumulators
        fill_f32<<<2048, 256, 0, stream>>>(hout[L], 0.f, (size_t)N * S);
        fill_f32<<<256, 256, 0, stream>>>(denom, 0.f, (size_t)N);
        fill_i32<<<256, 256, 0, stream>>>(emax, NEG_INF_KEY, (size_t)N);

        // xl = in @ Wl^T ; xr = in @ Wr^T  (f32 WMMA, fully unrolled, unguarded stores)
        if (L == 0) {
            wmma_linear_kernel<28, 112><<<gemmBlocks, 128, 0, stream>>>(lin[L], ldx[L], K, Wl[L], D, xl, N);
            wmma_linear_kernel<28, 112><<<gemmBlocks, 128, 0, stream>>>(lin[L], ldx[L], K, Wr[L], D, xr, N);
        } else if (L == 1) {
            wmma_linear_kernel<100, 64><<<gemmBlocks, 128, 0, stream>>>(lin[L], ldx[L], K, Wl[L], D, xl, N);
            wmma_linear_kernel<100, 64><<<gemmBlocks, 128, 0, stream>>>(lin[L], ldx[L], K, Wr[L], D, xr, N);
        } else {
            wmma_linear_kernel<60, 32><<<gemmBlocks, 128, 0, stream>>>(lin[L], ldx[L], K, Wl[L], D, xl, N);
            wmma_linear_kernel<60, 32><<<gemmBlocks, 128, 0, stream>>>(lin[L], ldx[L], K, Wr[L], D, xr, N);
        }

        edge_score_kernel<<<eblk, 256, 0, stream>>>(xl, xr, S, att[L], D,
                                                    srcA, dstA, E, ET, ebuf, emax);
        edge_aggregate_kernel<<<eblk, 256, 0, stream>>>(xl, S, srcA, dstA, E, ET,
                                                        ebuf, emax, denom, hout[L], D);
        node_finish_kernel<<<(N * D + 255) / 256, 256, 0, stream>>>(
            hout[L], S, D, bia[L], denom, N, (L < 2) ? 1 : 0);
    }

    // mean pool + MLP head
    fill_f32<<<64, 256, 0, stream>>>(pool, 0.f, (size_t)NUM_GRAPHS * 30);
    fill_f32<<<16, 256, 0, stream>>>(cnt, 0.f, (size_t)NUM_GRAPHS);
    pool_kernel<<<(N + 255) / 256, 256, 0, stream>>>(h3, 32, 30, batch, N, pool, cnt);
    mlp_kernel<<<(NUM_GRAPHS + 127) / 128, 128, 0, stream>>>(
        pool, cnt, Wfc1, bfc1, Wfc2, bfc2, (float*)d_out, NUM_GRAPHS);
}